// GINNet_pyg_59279138619791
// MI455X (gfx1250) — compile-verified
//
#include <hip/hip_runtime.h>
#include <hip/hip_bf16.h>

#define H_DIM 128
#define NB_STATS 256

typedef __attribute__((ext_vector_type(2))) float v2f;
typedef __attribute__((ext_vector_type(8))) float v8f;

// ---------------------------------------------------------------------------
// Tiled GEMM: out[N x 128] = preop(A)[N x 128] @ B[128 x 128] + bias
//   mode 0: preop = identity                      (embedding, A1 = X)
//   mode 1: preop = (1+eps[layer])*A1 + A2        (GIN aggregate)
//   mode 2: preop = relu(A1 * st[c] + st[128+c])  (folded BN+ReLU)
// One block = 16 rows x 128 cols; 8 waves, each wave one 16x16 WMMA tile.
// V_WMMA_F32_16X16X4_F32 fragment layouts (wave32):
//   A 16x4 : lanes 0-15 -> M=lane, {v0,v1}={K+0,K+1}; lanes 16-31 -> {K+2,K+3}
//   B 4x16 : lanes 0-15 -> N=lane, {v0,v1}={K+0,K+1}; lanes 16-31 -> {K+2,K+3}
//   C/D    : vgpr r -> row r (lanes 0-15) / row r+8 (lanes 16-31), N=lane&15
// ---------------------------------------------------------------------------
__global__ __launch_bounds__(256) void gin_gemm128(
    const float* __restrict__ A1, const float* __restrict__ A2,
    const float* __restrict__ B,  const float* __restrict__ bias,
    const float* __restrict__ st, const float* __restrict__ epsp,
    int layer, int mode, int nrows, float* __restrict__ out)
{
  __shared__ float As[16][132];            // pad 132: conflict-free column reads
  const int row0 = blockIdx.x << 4;
  const int tid  = threadIdx.x;

  float epsv = 1.0f;
  if (mode == 1) epsv = 1.0f + epsp[layer];

  // Stage A tile (with pre-op) into LDS: 2048 floats, coalesced.
  for (int i = tid; i < 16 * H_DIM; i += 256) {
    const int r = i >> 7, c = i & 127;
    const int row = row0 + r;
    float v = 0.0f;
    if (row < nrows) {
      v = A1[(size_t)row * H_DIM + c];
      if (mode == 1) {
        v = fmaf(epsv, v, A2[(size_t)row * H_DIM + c]);
      } else if (mode == 2) {
        v = fmaf(v, st[c], st[128 + c]);
        v = v > 0.0f ? v : 0.0f;
      }
    }
    As[r][c] = v;
  }
  __syncthreads();

  const int wave = tid >> 5;
  const int lane = tid & 31;
  const int hi   = lane >> 4;              // 0 or 1 (lane half)
  const int lm   = lane & 15;
  const int col  = (wave << 4) + lm;       // this wave's output column

  v8f acc = {};
  #pragma unroll
  for (int kk = 0; kk < H_DIM; kk += 4) {
    const int k0 = kk + 2 * hi;
    v2f a, b;
    a[0] = As[lm][k0];
    a[1] = As[lm][k0 + 1];
    b[0] = B[k0 * H_DIM + col];
    b[1] = B[(k0 + 1) * H_DIM + col];
    acc = __builtin_amdgcn_wmma_f32_16x16x4_f32(
        /*neg_a=*/false, a, /*neg_b=*/false, b,
        /*c_mod=*/(short)0, acc, /*reuse_a=*/false, /*reuse_b=*/false);
  }

  const float bv = bias[col];
  #pragma unroll
  for (int r = 0; r < 8; ++r) {
    const int row = row0 + r + 8 * hi;
    if (row < nrows) out[(size_t)row * H_DIM + col] = acc[r] + bv;
  }
}

// ---------------------------------------------------------------------------
// Edge scatter-add: agg[dst] += h[src].  One lane per (edge, 4 features).
// ---------------------------------------------------------------------------
__global__ __launch_bounds__(256) void gin_scatter(
    const float* __restrict__ h, const int* __restrict__ src,
    const int* __restrict__ dst, float* __restrict__ agg, int nedges)
{
  const int idx = blockIdx.x * 256 + threadIdx.x;
  const int e = idx >> 5;
  if (e >= nedges) return;
  const int q = (idx & 31) << 2;
  const int s = src[e], d = dst[e];
  const float4 v = *reinterpret_cast<const float4*>(h + (size_t)s * H_DIM + q);
  float* ap = agg + (size_t)d * H_DIM + q;
  unsafeAtomicAdd(ap + 0, v.x);
  unsafeAtomicAdd(ap + 1, v.y);
  unsafeAtomicAdd(ap + 2, v.z);
  unsafeAtomicAdd(ap + 3, v.w);
}

// ---------------------------------------------------------------------------
// Column stats pass 1: per-block partial sum / sumsq per column (deterministic).
// ---------------------------------------------------------------------------
__global__ __launch_bounds__(128) void gin_stats(
    const float* __restrict__ X, float* __restrict__ part, int nrows, int rpb)
{
  const int t  = threadIdx.x;              // column
  const int r0 = blockIdx.x * rpb;
  int r1 = r0 + rpb; if (r1 > nrows) r1 = nrows;
  float s = 0.0f, q = 0.0f;
  for (int r = r0; r < r1; ++r) {
    const float v = X[(size_t)r * H_DIM + t];
    s += v;
    q = fmaf(v, v, q);
  }
  part[blockIdx.x * 256 + t]       = s;
  part[blockIdx.x * 256 + 128 + t] = q;
}

// Pass 2: fold into BN (scale, shift): st[c]=g*rstd, st[128+c]=b-mean*scale.
__global__ __launch_bounds__(128) void gin_bnreduce(
    const float* __restrict__ part, int nb, const float* __restrict__ g,
    const float* __restrict__ b, int nrows, float* __restrict__ st)
{
  const int t = threadIdx.x;
  float s = 0.0f, q = 0.0f;
  for (int i = 0; i < nb; ++i) {
    s += part[i * 256 + t];
    q += part[i * 256 + 128 + t];
  }
  const float invn = 1.0f / (float)nrows;
  const float mean = s * invn;
  const float var  = fmaf(-mean, mean, q * invn);
  const float rstd = rsqrtf(var + 1e-5f);
  const float sc   = g[t] * rstd;
  st[t]       = sc;
  st[128 + t] = fmaf(-mean, sc, b[t]);
}

// X = relu(X * st[c] + st[128+c])   (in place)
__global__ __launch_bounds__(256) void gin_bnrelu(
    float* __restrict__ X, const float* __restrict__ st, int n)
{
  const int idx = blockIdx.x * 256 + threadIdx.x;
  if (idx >= n) return;
  const int c = idx & 127;
  float v = fmaf(X[idx], st[c], st[128 + c]);
  X[idx] = v > 0.0f ? v : 0.0f;
}

// h += relu(w * st[c] + st[128+c])   (residual, in place on h)
__global__ __launch_bounds__(256) void gin_residual(
    float* __restrict__ h, const float* __restrict__ w,
    const float* __restrict__ st, int n)
{
  const int idx = blockIdx.x * 256 + threadIdx.x;
  if (idx >= n) return;
  const int c = idx & 127;
  float v = fmaf(w[idx], st[c], st[128 + c]);
  v = v > 0.0f ? v : 0.0f;
  h[idx] += v;
}

// ---------------------------------------------------------------------------
// Prediction head: score[n][0..5] (+)= h[n] @ W[128x6] + b[6].
// One wave per node; lane l owns features 4l..4l+3; shuffle tree reduce.
// ---------------------------------------------------------------------------
__global__ __launch_bounds__(256) void gin_score(
    const float* __restrict__ h, const float* __restrict__ W,
    const float* __restrict__ b, float* __restrict__ score, int nrows, int init)
{
  const int wid  = (blockIdx.x * 256 + threadIdx.x) >> 5;
  const int lane = threadIdx.x & 31;
  if (wid >= nrows) return;
  const int f = lane << 2;
  const float4 hv = *reinterpret_cast<const float4*>(h + (size_t)wid * H_DIM + f);
  float acc[6];
  #pragma unroll
  for (int c = 0; c < 6; ++c) {
    acc[c] = hv.x * W[(f + 0) * 6 + c];
    acc[c] = fmaf(hv.y, W[(f + 1) * 6 + c], acc[c]);
    acc[c] = fmaf(hv.z, W[(f + 2) * 6 + c], acc[c]);
    acc[c] = fmaf(hv.w, W[(f + 3) * 6 + c], acc[c]);
  }
  #pragma unroll
  for (int off = 16; off > 0; off >>= 1) {
    #pragma unroll
    for (int c = 0; c < 6; ++c) acc[c] += __shfl_down(acc[c], off, 32);
  }
  if (lane == 0) {
    #pragma unroll
    for (int c = 0; c < 6; ++c) {
      const float v = acc[c] + b[c];
      score[(size_t)wid * 6 + c] = init ? v : (score[(size_t)wid * 6 + c] + v);
    }
  }
}

// ---------------------------------------------------------------------------
extern "C" void kernel_launch(void* const* d_in, const int* in_sizes, int n_in,
                              void* d_out, int out_size, void* d_ws, size_t ws_size,
                              hipStream_t stream) {
  const float* X     = (const float*)d_in[0];
  const int*   ei    = (const int*)  d_in[1];
  const float* emb_W = (const float*)d_in[3];
  const float* emb_b = (const float*)d_in[4];
  const float* eps   = (const float*)d_in[5];
  const float* W1    = (const float*)d_in[6];
  const float* b1    = (const float*)d_in[7];
  const float* g1    = (const float*)d_in[8];
  const float* be1   = (const float*)d_in[9];
  const float* W2    = (const float*)d_in[10];
  const float* b2    = (const float*)d_in[11];
  const float* ga    = (const float*)d_in[12];
  const float* ba    = (const float*)d_in[13];
  const float* go    = (const float*)d_in[14];
  const float* bo    = (const float*)d_in[15];
  const float* pW    = (const float*)d_in[16];
  const float* pb    = (const float*)d_in[17];
  float* score = (float*)d_out;

  const int N = in_sizes[0] / H_DIM;   // 50000
  const int E = in_sizes[1] / 2;       // 800000
  const int L = in_sizes[5];           // 4

  // Workspace layout (floats): h[N*128] | u[N*128] (agg/u/v/w) | partials | st
  float* h    = (float*)d_ws;
  float* u    = h + (size_t)N * H_DIM;
  float* part = u + (size_t)N * H_DIM;
  float* st   = part + (size_t)NB_STATS * 256;

  const int gemm_grid  = (N + 15) / 16;
  const int rpb        = (N + NB_STATS - 1) / NB_STATS;
  const int ew_grid    = (N * H_DIM + 255) / 256;
  const int score_grid = (N * 32 + 255) / 256;
  const int scat_grid  = (int)(((size_t)E * 32 + 255) / 256);

  // h = X @ emb_W + emb_b
  gin_gemm128<<<gemm_grid, 256, 0, stream>>>(X, nullptr, emb_W, emb_b,
                                             nullptr, nullptr, 0, 0, N, h);
  // score = h @ pred_W[0] + pred_b[0]
  gin_score<<<score_grid, 256, 0, stream>>>(h, pW, pb, score, N, 1);

  for (int l = 0; l < L; ++l) {
    // agg = segment_sum(h[src], dst)
    hipMemsetAsync(u, 0, (size_t)N * H_DIM * sizeof(float), stream);
    gin_scatter<<<scat_grid, 256, 0, stream>>>(h, ei, ei + E, u, E);
    // u = ((1+eps)*h + agg) @ W1 + b1      (in place over agg: safe per-row-slab)
    gin_gemm128<<<gemm_grid, 256, 0, stream>>>(h, u, W1 + (size_t)l * H_DIM * H_DIM,
                                               b1 + l * H_DIM, nullptr, eps, l, 1, N, u);
    // BN1 stats -> folded (scale, shift)
    gin_stats<<<NB_STATS, 128, 0, stream>>>(u, part, N, rpb);
    gin_bnreduce<<<1, 128, 0, stream>>>(part, NB_STATS, g1 + l * H_DIM, be1 + l * H_DIM, N, st);
    // u = relu(BN1(u)) @ W2 + b2           (BN+ReLU folded into A-load; in place)
    gin_gemm128<<<gemm_grid, 256, 0, stream>>>(u, nullptr, W2 + (size_t)l * H_DIM * H_DIM,
                                               b2 + l * H_DIM, st, nullptr, l, 2, N, u);
    // BN2 (ga,ba) stats -> apply relu(BN2) in place
    gin_stats<<<NB_STATS, 128, 0, stream>>>(u, part, N, rpb);
    gin_bnreduce<<<1, 128, 0, stream>>>(part, NB_STATS, ga + l * H_DIM, ba + l * H_DIM, N, st);
    gin_bnrelu<<<ew_grid, 256, 0, stream>>>(u, st, N * H_DIM);
    // BN3 (go,bo) stats -> h += relu(BN3(u))
    gin_stats<<<NB_STATS, 128, 0, stream>>>(u, part, N, rpb);
    gin_bnreduce<<<1, 128, 0, stream>>>(part, NB_STATS, go + l * H_DIM, bo + l * H_DIM, N, st);
    gin_residual<<<ew_grid, 256, 0, stream>>>(h, u, st, N * H_DIM);
    // score += h @ pred_W[l+1] + pred_b[l+1]
    gin_score<<<score_grid, 256, 0, stream>>>(h, pW + (size_t)(l + 1) * H_DIM * 6,
                                              pb + (l + 1) * 6, score, N, 0);
  }
}